// PureBiMambaScan_12378095747607
// MI455X (gfx1250) — compile-verified
//
#include <hip/hip_runtime.h>

typedef __attribute__((ext_vector_type(2))) float v2f;
typedef __attribute__((ext_vector_type(8))) float v8f;

#define N_ 4
#define S_ 2048
#define D_ 512
#define K_ 16
#define M_ (N_ * S_)   // 8192 flattened (n,s) rows

// ---------------------------------------------------------------------------
// softplus via hardware transcendentals: ln(1+e^x) = log2(1+2^(x*log2e))*ln2
// ---------------------------------------------------------------------------
__device__ __forceinline__ float softplus_f(float x) {
  float e  = __builtin_amdgcn_exp2f(x * 1.44269504088896340736f);
  float sp = __builtin_amdgcn_logf(1.0f + e) * 0.69314718055994530942f;
  return (x > 20.0f) ? x : sp;
}

// ---------------------------------------------------------------------------
// Phase 1: Out[M,Ncols] = Tm[M,Kdim] @ W[Ncols,Kdim]^T  (+bias, +softplus)
// One wave32 per 16x16 tile, V_WMMA_F32_16X16X4_F32 over the K dimension.
// A fragment (16x4): lane<16 holds row m0+l, K={k,k+1}; lane>=16 K={k+2,k+3}.
// B fragment (4x16): identical striping with W rows as columns.
// ---------------------------------------------------------------------------
__global__ __launch_bounds__(32) void wmma_gemm_sp(
    const float* __restrict__ Tm, const float* __restrict__ W,
    const float* __restrict__ bias, float* __restrict__ Out,
    int Kdim, int Ncols, int apply_sp) {
  const int m0   = blockIdx.x << 4;
  const int n0   = blockIdx.y << 4;
  const int lane = threadIdx.x;
  const int l    = lane & 15;
  const int hi   = lane >> 4;

  const float* ap = Tm + (size_t)(m0 + l) * Kdim + 2 * hi;
  const float* bp = W  + (size_t)(n0 + l) * Kdim + 2 * hi;

  v8f acc = {};
#pragma unroll 4
  for (int k = 0; k < Kdim; k += 4) {
    v2f av = *(const v2f*)(ap + k);
    v2f bv = *(const v2f*)(bp + k);
    // 8 args: (neg_a, A, neg_b, B, c_mod, C, reuse_a, reuse_b)
    acc = __builtin_amdgcn_wmma_f32_16x16x4_f32(false, av, false, bv,
                                                (short)0, acc, false, false);
  }

  float bs = bias ? bias[n0 + l] : 0.0f;
#pragma unroll
  for (int r = 0; r < 8; ++r) {
    int   row = r + (hi << 3);          // C/D layout: lanes16-31 hold M=r+8
    float v   = acc[r] + bs;
    if (apply_sp) v = softplus_f(v);
    Out[(size_t)(m0 + row) * Ncols + n0 + l] = v;
  }
}

// ---------------------------------------------------------------------------
// Phase 2: bidirectional selective scan.
// Thread = (n, d, k-group of 4, direction). 512 wave32s of independent
// recurrences; K-reduction deferred to one LDS reduce at the end.
// ---------------------------------------------------------------------------
__global__ __launch_bounds__(256) void bimamba_scan(
    const float* __restrict__ tokens, const float* __restrict__ log_A,
    const float* __restrict__ deltas, const float* __restrict__ Bs,
    const float* __restrict__ Cs, float* __restrict__ out) {
  const int tid  = threadIdx.x;
  const int dloc = tid & 63;    // contiguous d within wave -> coalesced loads
  const int kg   = tid >> 6;    // k-group constant per wave -> broadcast B/C
  const int d    = blockIdx.x * 64 + dloc;
  const int n    = blockIdx.y;
  const int dir  = blockIdx.z;
  const int k0   = kg << 2;

  const float LOG2E = 1.44269504088896340736f;
  // A2 = A_cont * log2e  so the per-step exp is a single v_exp_f32
  float A2[4];
#pragma unroll
  for (int j = 0; j < 4; ++j)
    A2[j] = -__builtin_amdgcn_exp2f(log_A[d * K_ + k0 + j] * LOG2E) * LOG2E;

  float h0 = 0.f, h1 = 0.f, h2 = 0.f, h3 = 0.f, acc = 0.f;

  long       te    = dir ? (long)(S_ - 1) : 0;
  const long dt    = dir ? -1 : 1;
  const size_t nb  = (size_t)n * S_;

  for (int t = 0; t < S_; ++t, te += dt) {
    const size_t row = nb + (size_t)te;
    float  delta = deltas[row * D_ + d];
    float  x     = tokens[row * D_ + d];
    float4 Bv    = *(const float4*)(Bs + row * K_ + k0);
    float4 Cv    = *(const float4*)(Cs + row * K_ + k0);
    float  bx    = delta * x;

    float a0 = __builtin_amdgcn_exp2f(delta * A2[0]);
    float a1 = __builtin_amdgcn_exp2f(delta * A2[1]);
    float a2 = __builtin_amdgcn_exp2f(delta * A2[2]);
    float a3 = __builtin_amdgcn_exp2f(delta * A2[3]);

    h0 = fmaf(a0, h0, Bv.x * bx);
    h1 = fmaf(a1, h1, Bv.y * bx);
    h2 = fmaf(a2, h2, Bv.z * bx);
    h3 = fmaf(a3, h3, Bv.w * bx);

    acc = fmaf(h0, Cv.x, acc);
    acc = fmaf(h1, Cv.y, acc);
    acc = fmaf(h2, Cv.z, acc);
    acc = fmaf(h3, Cv.w, acc);
  }

  __shared__ float red[256];
  red[tid] = acc;
  __syncthreads();
  if (tid < 64) {   // here d == blockIdx.x*64 + tid
    float s = red[tid] + red[tid + 64] + red[tid + 128] + red[tid + 192];
    // exactly two adders (fwd,bwd) per element: order-independent fp32 result
    atomicAdd(out + (size_t)n * D_ + blockIdx.x * 64 + tid, s * (1.0f / (float)S_));
  }
}

__global__ void zero_f32(float* __restrict__ p, int n) {
  int i = blockIdx.x * blockDim.x + threadIdx.x;
  if (i < n) p[i] = 0.0f;
}

// ---------------------------------------------------------------------------
extern "C" void kernel_launch(void* const* d_in, const int* in_sizes, int n_in,
                              void* d_out, int out_size, void* d_ws, size_t ws_size,
                              hipStream_t stream) {
  const float* tokens  = (const float*)d_in[0];  // [N,S,D]
  const float* log_A   = (const float*)d_in[1];  // [D,K]
  const float* W_B     = (const float*)d_in[2];  // [K,D]
  const float* W_C     = (const float*)d_in[3];  // [K,D]
  const float* W_delta = (const float*)d_in[4];  // [D,D]
  const float* b_delta = (const float*)d_in[5];  // [D]
  float* out = (float*)d_out;                    // [N,D]

  // workspace layout (fp32): deltas[M,D] | Bs[M,K] | Cs[M,K]  (~17 MB)
  float* deltas = (float*)d_ws;
  float* Bsbuf  = deltas + (size_t)M_ * D_;
  float* Csbuf  = Bsbuf  + (size_t)M_ * K_;

  // Phase 1: WMMA GEMMs (softplus+bias fused into the deltas store)
  wmma_gemm_sp<<<dim3(M_ / 16, D_ / 16), 32, 0, stream>>>(
      tokens, W_delta, b_delta, deltas, D_, D_, 1);
  wmma_gemm_sp<<<dim3(M_ / 16, 1), 32, 0, stream>>>(
      tokens, W_B, nullptr, Bsbuf, D_, K_, 0);
  wmma_gemm_sp<<<dim3(M_ / 16, 1), 32, 0, stream>>>(
      tokens, W_C, nullptr, Csbuf, D_, K_, 0);

  // Phase 2: zero output, then bidirectional scan (fwd/bwd as blockIdx.z)
  zero_f32<<<(N_ * D_ + 255) / 256, 256, 0, stream>>>(out, N_ * D_);
  bimamba_scan<<<dim3(D_ / 64, N_, 2), 256, 0, stream>>>(
      tokens, log_A, deltas, Bsbuf, Csbuf, out);
}